// IngredientGNN_75247827026326
// MI455X (gfx1250) — compile-verified
//
#include <hip/hip_runtime.h>
#include <hip/hip_bf16.h>

typedef __attribute__((ext_vector_type(2))) float v2f;
typedef __attribute__((ext_vector_type(8))) float v8f;

#define BN_EPS 1e-5f

// ---------------------------------------------------------------------------
// Degree / normalization precompute
// ---------------------------------------------------------------------------
__global__ void k_deg_init(unsigned* __restrict__ deg, int N) {
    int i = blockIdx.x * 256 + threadIdx.x;
    if (i < N) deg[i] = 1u;  // self-loop
}

__global__ void k_deg_accum(const int* __restrict__ dst, unsigned* __restrict__ deg,
                            long long E) {
    long long e = (long long)blockIdx.x * 256 + threadIdx.x;
    if (e < E) atomicAdd(&deg[dst[e]], 1u);
}

__global__ void k_dinv(float* __restrict__ buf, int N) {
    int i = blockIdx.x * 256 + threadIdx.x;
    if (i < N) {
        unsigned u = ((unsigned*)buf)[i];
        buf[i] = rsqrtf((float)u);   // deg >= 1 always (self-loops)
    }
}

__global__ void k_edge_norm(const int* __restrict__ src, const int* __restrict__ dst,
                            const float* __restrict__ dinv, float* __restrict__ norm,
                            long long E) {
    long long e = (long long)blockIdx.x * 256 + threadIdx.x;
    if (e < E) norm[e] = dinv[src[e]] * dinv[dst[e]];
}

// ---------------------------------------------------------------------------
// Dense transform h = X @ W  (N x 64) @ (64 x 64), fp32 WMMA 16x16x4.
// One wave computes a 16-row x 64-col slab; W staged in LDS.
// A frag (16x4 f32): lanes 0-15 -> M=lane, K=k..k+1; lanes 16-31 -> K=k+2..k+3.
// B frag (4x16 f32): mirrored K split across lane halves, N = lane&15.
// C/D (16x16 f32): VGPR v -> M = v + 8*(lane>=16), N = lane&15.
// ---------------------------------------------------------------------------
__global__ void __launch_bounds__(256) k_gemm64_wmma(const float* __restrict__ X,
                                                     const float* __restrict__ W,
                                                     float* __restrict__ H, int N) {
    __shared__ float Ws[64 * 64];
    int tx = threadIdx.x;
    #pragma unroll
    for (int i = 0; i < 16; ++i) Ws[tx + i * 256] = W[tx + i * 256];
    __syncthreads();

    int wave = tx >> 5;
    int lane = tx & 31;
    int half = lane >> 4;     // 0: K lo pair, 1: K hi pair
    int l    = lane & 15;

    int base = (blockIdx.x * 8 + wave) * 16;
    if (base >= N) return;    // wave-uniform exit -> EXEC stays all-ones for WMMA

    const float* xrow = X + (size_t)(base + l) * 64;
    v8f acc0 = {}, acc1 = {}, acc2 = {}, acc3 = {};

    #pragma unroll
    for (int k = 0; k < 64; k += 4) {
        int kk = k + half * 2;
        v2f a;
        a.x = xrow[kk];
        a.y = xrow[kk + 1];
        const float* w0 = &Ws[(size_t)kk * 64];
        const float* w1 = &Ws[(size_t)(kk + 1) * 64];
        v2f b0, b1, b2, b3;
        b0.x = w0[l];      b0.y = w1[l];
        b1.x = w0[16 + l]; b1.y = w1[16 + l];
        b2.x = w0[32 + l]; b2.y = w1[32 + l];
        b3.x = w0[48 + l]; b3.y = w1[48 + l];
        acc0 = __builtin_amdgcn_wmma_f32_16x16x4_f32(false, a, false, b0, (short)0, acc0, false, false);
        acc1 = __builtin_amdgcn_wmma_f32_16x16x4_f32(false, a, false, b1, (short)0, acc1, false, false);
        acc2 = __builtin_amdgcn_wmma_f32_16x16x4_f32(false, a, false, b2, (short)0, acc2, false, false);
        acc3 = __builtin_amdgcn_wmma_f32_16x16x4_f32(false, a, false, b3, (short)0, acc3, false, false);
    }

    float* hout = H + (size_t)base * 64;
    #pragma unroll
    for (int v = 0; v < 8; ++v) {
        size_t r = (size_t)(v + half * 8) * 64;
        hout[r + l]      = acc0[v];
        hout[r + 16 + l] = acc1[v];
        hout[r + 32 + l] = acc2[v];
        hout[r + 48 + l] = acc3[v];
    }
}

// ---------------------------------------------------------------------------
// out[i,:] = h[i,:] * dinv[i]^2 + b   (self-loop contribution + bias)
// ---------------------------------------------------------------------------
__global__ void k_agg_init(const float* __restrict__ H, const float* __restrict__ dinv,
                           const float* __restrict__ b, float* __restrict__ out,
                           long long total) {
    long long i = (long long)blockIdx.x * 256 + threadIdx.x;
    if (i >= total) return;
    int node = (int)(i >> 6);
    int d    = (int)(i & 63);
    float di = dinv[node];
    out[i] = H[i] * di * di + b[d];
}

// ---------------------------------------------------------------------------
// Edge scatter: 16 lanes per edge, each handles 4 consecutive features.
// fp32 atomics resolve in L2 (all hot buffers fit in 192MB L2).
// ---------------------------------------------------------------------------
__global__ void k_scatter(const float* __restrict__ H, const int* __restrict__ src,
                          const int* __restrict__ dst, const float* __restrict__ norm,
                          float* __restrict__ out, long long E) {
    long long t = (long long)blockIdx.x * 256 + threadIdx.x;
    long long e = t >> 4;
    if (e >= E) return;
    int q = (int)(t & 15) * 4;
    int s = src[e], d = dst[e];
    float n = norm[e];
    const float4 v = *(const float4*)(H + (size_t)s * 64 + q);
    float* o = out + (size_t)d * 64 + q;
    unsafeAtomicAdd(o + 0, v.x * n);
    unsafeAtomicAdd(o + 1, v.y * n);
    unsafeAtomicAdd(o + 2, v.z * n);
    unsafeAtomicAdd(o + 3, v.w * n);
}

// ---------------------------------------------------------------------------
// BatchNorm statistics: per-feature sum and sum of squares.
// 256 threads = 4 row-subsets x 64 features; coalesced row reads.
// ---------------------------------------------------------------------------
__global__ void k_zero128(float* __restrict__ p) {
    p[threadIdx.x] = 0.0f;  // 128 threads: sum[64] + sumsq[64]
}

__global__ void __launch_bounds__(256) k_bn_stats(const float* __restrict__ X,
                                                  float* __restrict__ sum,
                                                  float* __restrict__ sumsq, int N) {
    __shared__ float ls[256], lq[256];
    int tx  = threadIdx.x;
    int d   = tx & 63;
    int sub = tx >> 6;
    float s = 0.0f, q = 0.0f;
    #pragma unroll 4
    for (int i = 0; i < 64; ++i) {
        int r = blockIdx.x * 256 + sub + i * 4;
        if (r < N) {
            float v = X[(size_t)r * 64 + d];
            s += v;
            q += v * v;
        }
    }
    ls[tx] = s; lq[tx] = q;
    __syncthreads();
    if (tx < 64) {
        s = ls[tx] + ls[tx + 64] + ls[tx + 128] + ls[tx + 192];
        q = lq[tx] + lq[tx + 64] + lq[tx + 128] + lq[tx + 192];
        unsafeAtomicAdd(&sum[tx], s);
        unsafeAtomicAdd(&sumsq[tx], q);
    }
}

__global__ void k_bn_relu(const float* __restrict__ X, const float* __restrict__ sum,
                          const float* __restrict__ sumsq, const float* __restrict__ g,
                          const float* __restrict__ be, float* __restrict__ Y,
                          long long total, float invN) {
    long long i = (long long)blockIdx.x * 256 + threadIdx.x;
    if (i >= total) return;
    int d = (int)(i & 63);
    float m   = sum[d] * invN;
    float var = sumsq[d] * invN - m * m;
    float y = g[d] * (X[i] - m) * rsqrtf(var + BN_EPS) + be[d];
    Y[i] = fmaxf(y, 0.0f);
}

// ---------------------------------------------------------------------------
// Host-side orchestration
// ---------------------------------------------------------------------------
static inline int cdiv_ll(long long a, long long b) { return (int)((a + b - 1) / b); }

extern "C" void kernel_launch(void* const* d_in, const int* in_sizes, int n_in,
                              void* d_out, int out_size, void* d_ws, size_t ws_size,
                              hipStream_t stream) {
    const float* x   = (const float*)d_in[0];
    const int*   ei  = (const int*)d_in[1];
    const float* W1  = (const float*)d_in[2];
    const float* b1  = (const float*)d_in[3];
    const float* g1  = (const float*)d_in[4];
    const float* be1 = (const float*)d_in[5];
    const float* W2  = (const float*)d_in[6];
    const float* b2  = (const float*)d_in[7];
    const float* g2  = (const float*)d_in[8];
    const float* be2 = (const float*)d_in[9];
    const float* W3  = (const float*)d_in[10];
    const float* b3  = (const float*)d_in[11];

    const long long N = in_sizes[0] / 64;
    const long long E = in_sizes[1] / 2;
    const int* src = ei;
    const int* dst = ei + E;

    float* ws    = (float*)d_ws;
    float* dinv  = ws;                 // N
    float* norm  = dinv + N;           // E
    float* h     = norm + E;           // N*64
    float* agg   = h + N * 64;         // N*64
    float* xbuf  = agg + N * 64;       // N*64
    float* sum   = xbuf + N * 64;      // 64
    float* sumsq = sum + 64;           // 64

    float* outp = (float*)d_out;
    const float invN = 1.0f / (float)N;
    const long long ND = N * 64;

    // normalization precompute
    k_deg_init<<<cdiv_ll(N, 256), 256, 0, stream>>>((unsigned*)dinv, (int)N);
    k_deg_accum<<<cdiv_ll(E, 256), 256, 0, stream>>>(dst, (unsigned*)dinv, E);
    k_dinv<<<cdiv_ll(N, 256), 256, 0, stream>>>(dinv, (int)N);
    k_edge_norm<<<cdiv_ll(E, 256), 256, 0, stream>>>(src, dst, dinv, norm, E);

    const int gemm_grid    = cdiv_ll(N, 128);       // 8 waves/block * 16 rows
    const int elem_grid    = cdiv_ll(ND, 256);
    const int scatter_grid = cdiv_ll(E * 16, 256);
    const int stats_grid   = cdiv_ll(N, 256);

    // ---- layer 1 ----
    k_gemm64_wmma<<<gemm_grid, 256, 0, stream>>>(x, W1, h, (int)N);
    k_agg_init<<<elem_grid, 256, 0, stream>>>(h, dinv, b1, agg, ND);
    k_scatter<<<scatter_grid, 256, 0, stream>>>(h, src, dst, norm, agg, E);
    k_zero128<<<1, 128, 0, stream>>>(sum);
    k_bn_stats<<<stats_grid, 256, 0, stream>>>(agg, sum, sumsq, (int)N);
    k_bn_relu<<<elem_grid, 256, 0, stream>>>(agg, sum, sumsq, g1, be1, xbuf, ND, invN);

    // ---- layer 2 ----
    k_gemm64_wmma<<<gemm_grid, 256, 0, stream>>>(xbuf, W2, h, (int)N);
    k_agg_init<<<elem_grid, 256, 0, stream>>>(h, dinv, b2, agg, ND);
    k_scatter<<<scatter_grid, 256, 0, stream>>>(h, src, dst, norm, agg, E);
    k_zero128<<<1, 128, 0, stream>>>(sum);
    k_bn_stats<<<stats_grid, 256, 0, stream>>>(agg, sum, sumsq, (int)N);
    k_bn_relu<<<elem_grid, 256, 0, stream>>>(agg, sum, sumsq, g2, be2, xbuf, ND, invN);

    // ---- layer 3 (no BN/ReLU, write straight to d_out) ----
    k_gemm64_wmma<<<gemm_grid, 256, 0, stream>>>(xbuf, W3, h, (int)N);
    k_agg_init<<<elem_grid, 256, 0, stream>>>(h, dinv, b3, outp, ND);
    k_scatter<<<scatter_grid, 256, 0, stream>>>(h, src, dst, norm, outp, E);
}